// ResidualSFDIVEQ_53017076302227
// MI455X (gfx1250) — compile-verified
//
#include <hip/hip_runtime.h>
#include <hip/hip_bf16.h>
#include <math.h>

// Problem constants (match reference)
#define NN 65536
#define DD 64
#define CC 4
#define KK 1024

#define ROW_STRIDE 68  // 64 + 4 dword pad: bank-conflict-free LDS rows, 16B aligned

typedef float v2f __attribute__((ext_vector_type(2)));
typedef float v4f __attribute__((ext_vector_type(4)));
typedef float v8f __attribute__((ext_vector_type(8)));

// ---------------------------------------------------------------------------
// Zero the histogram workspace (graph-capture safe)
// ---------------------------------------------------------------------------
__global__ void zero_counts_kernel(int* __restrict__ counts) {
    int t = blockIdx.x * blockDim.x + threadIdx.x;
    if (t < CC * KK) counts[t] = 0;
}

// ---------------------------------------------------------------------------
// Precompute ||codeword||^2 for all C*K codewords
// ---------------------------------------------------------------------------
__global__ void cnorm_kernel(const float* __restrict__ cb, float* __restrict__ cnorm) {
    int t = blockIdx.x * blockDim.x + threadIdx.x;  // 0 .. C*K-1
    if (t >= CC * KK) return;
    const float* p = cb + (size_t)t * DD;
    float s = 0.f;
#pragma unroll
    for (int d = 0; d < DD; d += 4) {
        v4f v = *(const v4f*)(p + d);
        s += v.x * v.x + v.y * v.y + v.z * v.z + v.w * v.w;
    }
    cnorm[t] = s;
}

// ---------------------------------------------------------------------------
// Main residual-VQ kernel.
// Block: 256 threads = 8 wave32. Wave w owns rows [blk*128 + w*16, +16).
// ---------------------------------------------------------------------------
__global__ __launch_bounds__(256)
void rvq_main_kernel(const float* __restrict__ z,
                     const float* __restrict__ cb,
                     const float* __restrict__ noise,
                     const float* __restrict__ cnorm,
                     int*   __restrict__ counts,
                     float* __restrict__ out_zq,
                     float* __restrict__ out_idx) {
    __shared__ float cb_ld[64 * ROW_STRIDE];        // 64 staged codewords
    __shared__ float cn_ld[64];                     // their squared norms
    __shared__ float rem_ld[8 * 16 * ROW_STRIDE];   // per-wave residual tiles
    __shared__ int   idx_ld[8 * 16];                // per-wave chosen indices

    const int tid  = threadIdx.x;
    const int wave = tid >> 5;
    const int lane = tid & 31;
    const int l15  = lane & 15;
    const int hi   = lane >> 4;          // 0 (lanes 0-15) or 1 (lanes 16-31)
    const int rowbase = blockIdx.x * 128 + wave * 16;

    float* remw = &rem_ld[wave * 16 * ROW_STRIDE];
    int*   idxw = &idx_ld[wave * 16];

    // staging decomposition for codebook chunks (64 codewords x 64 dims)
    const int srow = tid >> 2;           // codeword row this thread stages
    const int squad = (tid & 3) * 16;    // 16-dim quarter

    // --- stage z rows into the wave's residual tile (2 rows / iter, b128) ---
#pragma unroll
    for (int i = 0; i < 8; ++i) {
        int r  = 2 * i + hi;
        int cq = l15 * 4;
        v4f v = *(const v4f*)(z + (size_t)(rowbase + r) * DD + cq);
        *(v4f*)(remw + r * ROW_STRIDE + cq) = v;
    }

    for (int c = 0; c < CC; ++c) {
        __syncthreads();  // rem tile (cross-lane LDS) settled; cb_ld reusable

        // --- A fragments: 16x64 residual as 16 fp32 WMMA A-tiles (v2f/lane).
        // ISA layout 16x4 fp32 A: lanes 0-15 -> M=lane, K={0,1};
        //                         lanes 16-31 -> M=lane-16, K={2,3}.
        v2f a[16];
#pragma unroll
        for (int t = 0; t < 16; ++t)
            a[t] = *(const v2f*)(remw + l15 * ROW_STRIDE + 4 * t + 2 * hi);

        // --- R = ||rem_row||^2 (row-constant term, included to mimic the
        // reference's fp32 rounding in d2 = (R - 2*dot) + cn).
        float rloc = 0.f;
#pragma unroll
        for (int t = 0; t < 16; ++t)
            rloc += a[t].x * a[t].x + a[t].y * a[t].y;
        rloc += __shfl_xor(rloc, 16, 32);       // lane L now holds R[row l15]
        float Rj[8];
#pragma unroll
        for (int j = 0; j < 8; ++j)
            Rj[j] = __shfl(rloc, hi * 8 + j, 32);  // R for this lane's 8 rows

        float minv[8];
        int   mini[8];
#pragma unroll
        for (int j = 0; j < 8; ++j) { minv[j] = 3.4e38f; mini[j] = 0; }

        const float* cbc = cb + (size_t)c * KK * DD;

        // --- register prefetch of chunk 0 ---
        v4f pf0, pf1, pf2, pf3;
        float cnpf = 0.f;
        {
            const float* src = cbc + (size_t)srow * DD + squad;
            pf0 = *(const v4f*)(src + 0);
            pf1 = *(const v4f*)(src + 4);
            pf2 = *(const v4f*)(src + 8);
            pf3 = *(const v4f*)(src + 12);
            if (tid < 64) cnpf = cnorm[c * KK + tid];
        }

        for (int chunk = 0; chunk < KK / 64; ++chunk) {
            __syncthreads();  // all waves done reading previous chunk
            {
                float* dst = cb_ld + srow * ROW_STRIDE + squad;
                *(v4f*)(dst + 0)  = pf0;
                *(v4f*)(dst + 4)  = pf1;
                *(v4f*)(dst + 8)  = pf2;
                *(v4f*)(dst + 12) = pf3;
                if (tid < 64) cn_ld[tid] = cnpf;
            }
            if (chunk + 1 < KK / 64) {  // prefetch next chunk (hidden by compute)
                const float* src = cbc + (size_t)((chunk + 1) * 64 + srow) * DD + squad;
                pf0 = *(const v4f*)(src + 0);
                pf1 = *(const v4f*)(src + 4);
                pf2 = *(const v4f*)(src + 8);
                pf3 = *(const v4f*)(src + 12);
                if (tid < 64) cnpf = cnorm[c * KK + (chunk + 1) * 64 + tid];
            }
            __syncthreads();

            // two k-tiles in flight: independent WMMA chains sharing A
            for (int kt = 0; kt < 4; kt += 2) {
                v8f acc0 = {0.f, 0.f, 0.f, 0.f, 0.f, 0.f, 0.f, 0.f};
                v8f acc1 = {0.f, 0.f, 0.f, 0.f, 0.f, 0.f, 0.f, 0.f};
                const float* b0 = cb_ld + (kt * 16 + l15) * ROW_STRIDE + 2 * hi;
                const float* b1 = b0 + 16 * ROW_STRIDE;
#pragma unroll
                for (int t = 0; t < 16; ++t) {
                    v2f bb0 = *(const v2f*)(b0 + 4 * t);
                    v2f bb1 = *(const v2f*)(b1 + 4 * t);
                    acc0 = __builtin_amdgcn_wmma_f32_16x16x4_f32(
                        false, a[t], false, bb0, (short)0, acc0, false, false);
                    acc1 = __builtin_amdgcn_wmma_f32_16x16x4_f32(
                        false, a[t], false, bb1, (short)0, acc1, false, false);
                }
                // D layout: acc[j] = dot; lanes<16 -> rows j, lanes>=16 -> rows j+8
                float cn0 = cn_ld[kt * 16 + l15];
                float cn1 = cn_ld[kt * 16 + 16 + l15];
                int k0 = chunk * 64 + kt * 16 + l15;
#pragma unroll
                for (int j = 0; j < 8; ++j) {
                    // (R - 2*dot) + cn with the reference's association/rounding
                    float s0 = __fadd_rn(__fsub_rn(Rj[j], 2.0f * acc0[j]), cn0);
                    if (s0 < minv[j]) { minv[j] = s0; mini[j] = k0; }
                    float s1 = __fadd_rn(__fsub_rn(Rj[j], 2.0f * acc1[j]), cn1);
                    if (s1 < minv[j]) { minv[j] = s1; mini[j] = k0 + 16; }
                }
            }
        }

        // --- cross-lane argmin within each 16-lane half (ties -> lower idx)
#pragma unroll
        for (int j = 0; j < 8; ++j) {
            float v = minv[j];
            int   ix = mini[j];
#pragma unroll
            for (int off = 1; off < 16; off <<= 1) {
                float ov = __shfl_xor(v, off, 32);
                int   oi = __shfl_xor(ix, off, 32);
                if (ov < v || (ov == v && oi < ix)) { v = ov; ix = oi; }
            }
            minv[j] = v; mini[j] = ix;
        }

        // lane 0 holds rows 0..7, lane 16 holds rows 8..15
        if (l15 == 0) {
#pragma unroll
            for (int j = 0; j < 8; ++j) {
                int row = hi * 8 + j;
                int ix  = mini[j];
                idxw[row] = ix;
                out_idx[(size_t)c * NN + rowbase + row] = (float)ix;
                atomicAdd(&counts[c * KK + ix], 1);
            }
        }
        __syncthreads();  // publish idxw to the wave (and fence cb_ld stage)

        // --- rem -= chosen codeword (L2-hot gather; 2 cols / lane / row) ---
        for (int r = 0; r < 16; ++r) {
            int ix = idxw[r];
            v2f q   = *(const v2f*)(cbc + (size_t)ix * DD + lane * 2);
            v2f cur = *(v2f*)(remw + r * ROW_STRIDE + lane * 2);
            cur.x -= q.x; cur.y -= q.y;
            *(v2f*)(remw + r * ROW_STRIDE + lane * 2) = cur;
        }
    }

    // --- epilogue: z_q = z + ||rem|| * rv / max(||rv||, eps),
    //     rv = 1e-3*noise - rem   (direction = z_hard - z = -rem_final)
    for (int r = 0; r < 16; ++r) {
        int grow = rowbase + r;
        v2f rem2 = *(const v2f*)(remw + r * ROW_STRIDE + lane * 2);
        v2f nz   = *(const v2f*)(noise + (size_t)grow * DD + lane * 2);
        float rvx = 1e-3f * nz.x - rem2.x;
        float rvy = 1e-3f * nz.y - rem2.y;
        float p_rv  = rvx * rvx + rvy * rvy;
        float p_err = rem2.x * rem2.x + rem2.y * rem2.y;
#pragma unroll
        for (int off = 1; off < 32; off <<= 1) {
            p_rv  += __shfl_xor(p_rv, off, 32);
            p_err += __shfl_xor(p_err, off, 32);
        }
        float scale = sqrtf(p_err) / fmaxf(sqrtf(p_rv), 1e-12f);
        v2f zz = *(const v2f*)(z + (size_t)grow * DD + lane * 2);
        v2f o;
        o.x = zz.x + scale * rvx;
        o.y = zz.y + scale * rvy;
        *(v2f*)(out_zq + (size_t)grow * DD + lane * 2) = o;
    }
}

// ---------------------------------------------------------------------------
// Histogram -> perplexity per codebook
// ---------------------------------------------------------------------------
__global__ __launch_bounds__(256)
void perp_kernel(const int* __restrict__ counts, float* __restrict__ out_perp) {
    __shared__ float red[256];
    int c = blockIdx.x;
    float s = 0.f;
    for (int k = threadIdx.x; k < KK; k += 256) {
        float p = (float)counts[c * KK + k] * (1.0f / (float)NN);
        s += p * logf(p + 1e-10f);
    }
    red[threadIdx.x] = s;
    __syncthreads();
    for (int st = 128; st > 0; st >>= 1) {
        if (threadIdx.x < st) red[threadIdx.x] += red[threadIdx.x + st];
        __syncthreads();
    }
    if (threadIdx.x == 0) out_perp[c] = expf(-red[0]);
}

// ---------------------------------------------------------------------------
extern "C" void kernel_launch(void* const* d_in, const int* in_sizes, int n_in,
                              void* d_out, int out_size, void* d_ws, size_t ws_size,
                              hipStream_t stream) {
    const float* z     = (const float*)d_in[0];  // [N, D]
    const float* cb    = (const float*)d_in[1];  // [C, K, D]
    const float* noise = (const float*)d_in[2];  // [N, D]

    float* out      = (float*)d_out;
    float* out_zq   = out;                                      // N*D
    float* out_idx  = out + (size_t)NN * DD;                    // C*N (as float)
    float* out_perp = out + (size_t)NN * DD + (size_t)CC * NN;  // C

    int*   counts = (int*)d_ws;                                 // C*K ints
    float* cnorm  = (float*)((char*)d_ws + (size_t)CC * KK * sizeof(int));

    zero_counts_kernel<<<(CC * KK + 255) / 256, 256, 0, stream>>>(counts);
    cnorm_kernel<<<(CC * KK + 255) / 256, 256, 0, stream>>>(cb, cnorm);
    rvq_main_kernel<<<NN / 128, 256, 0, stream>>>(z, cb, noise, cnorm,
                                                  counts, out_zq, out_idx);
    perp_kernel<<<CC, 256, 0, stream>>>(counts, out_perp);
}